// LocallyConnected2D_77137612636308
// MI455X (gfx1250) — compile-verified
//
#include <hip/hip_runtime.h>

typedef float v2f __attribute__((ext_vector_type(2)));
typedef float v8f __attribute__((ext_vector_type(8)));

#define BATCH 32
#define C_IN 128
#define H_IN 64
#define W_IN 64
#define HW_IN (H_IN * W_IN)          // 4096
#define C_OUT 256
#define H_OUT 62
#define W_OUT 62
#define PQ (H_OUT * W_OUT)           // 3844
#define M_TOTAL (BATCH * PQ)         // 123008
#define K_W (C_IN * 9)               // 1152 (weight k-stride per o)

#define M_BLK 128
#define N_BLK 128
#define K_C 32                        // channels per K chunk (K order: rs-major)
#define LDA 34                        // padded LDS pitch (dwords) for As/Bs
#define LDD 65                        // padded LDS pitch for D transpose
#define STAGE_F (2 * M_BLK * LDA)     // floats per stage buffer (As+Bs = 8704)
#define NCHUNK 36                     // 9 rs * 4 c-blocks

// Implicit-GEMM 3x3 VALID conv, f32, V_WMMA_F32_16X16X4_F32, double-buffered
// LDS software pipeline: global loads for chunk t+1 issue before the 64 WMMAs
// of chunk t, ds_stores land in the alternate buffer, one barrier per chunk.
__global__ __launch_bounds__(256) void conv3x3_wmma_f32_kernel(
    const float* __restrict__ x,   // [32,128,64,64]
    const float* __restrict__ w,   // [256,128,3,3]
    float* __restrict__ out)       // [32,256,62,62]
{
    // 2 stage buffers (69,632 B); epilogue D (8 waves * 32*65 = 66,560 B) aliases.
    __shared__ float smem[2 * STAGE_F];

    const int tid  = threadIdx.x;
    const int lane = tid & 31;
    const int wid  = tid >> 5;
    const int hi   = lane >> 4;          // upper half-wave -> upper K pair / M+8
    const int ln   = lane & 15;

    const int m0 = blockIdx.x * M_BLK;
    const int n0 = blockIdx.y * N_BLK;

    const int wm = (wid & 3) * 32;       // wave M offset: 0,32,64,96
    const int wn = (wid >> 2) * 64;      // wave N offset: 0,64

    // ---- A staging geometry: element e = tid + 256*j -> row i = tid&127 (fixed),
    // kk = (tid>>7) + 2*j. Per-thread base + j*2*HW_IN global, base + 2*j LDS.
    const int ai   = tid & 127;
    const int akk0 = tid >> 7;           // 0 or 1
    int am  = m0 + ai;
    int ab  = am / PQ;
    int apq = am - ab * PQ;
    int ap  = apq / W_OUT;
    int aq  = apq - ap * W_OUT;
    const int agbase = ((ab * C_IN + akk0) * H_IN + ap) * W_IN + aq; // c=akk0,kh=kw=0
    const int asbase = ai * LDA + akk0;

    // ---- B staging geometry: kk = tid&31 (fixed), n = (tid>>5) + 8*j.
    const int bkk    = tid & 31;
    const int bn     = tid >> 5;         // 0..7
    const int bgbase = (n0 + bn) * K_W + bkk * 9;   // + c0*9 + rs + j*8*K_W
    const int bsbase = bn * LDA + bkk;              // + j*8*LDA

    // ---- fragment LDS bases (dword indices); all even -> ds_load_b64 legal.
    const int afrag = (wm + ln) * LDA + 2 * hi;     // + mt*16*LDA + kk
    const int bfrag = (wn + ln) * LDA + 2 * hi;     // + nt*16*LDA + kk

    v8f acc[2][4];
    #pragma unroll
    for (int mt = 0; mt < 2; ++mt)
        #pragma unroll
        for (int nt = 0; nt < 4; ++nt)
            acc[mt][nt] = (v8f){};

    float ar[16], br[16];   // in-flight staging registers for the next chunk

    // chunk t: rs = t>>2, c0 = (t&3)*K_C
    auto stage_load = [&](int t) {
        int rs = t >> 2;
        int c0 = (t & 3) * K_C;
        int kh = rs / 3;
        int kw = rs - kh * 3;
        int aoff = agbase + kh * W_IN + kw + c0 * HW_IN;
        int boff = bgbase + c0 * 9 + rs;
        #pragma unroll
        for (int j = 0; j < 16; ++j)
            ar[j] = x[aoff + j * (2 * HW_IN)];
        #pragma unroll
        for (int j = 0; j < 16; ++j)
            br[j] = w[boff + j * (8 * K_W)];
    };

    auto stage_store = [&](int buf) {
        float* As = smem + buf * STAGE_F;
        float* Bs = As + M_BLK * LDA;
        #pragma unroll
        for (int j = 0; j < 16; ++j)
            As[asbase + 2 * j] = ar[j];
        #pragma unroll
        for (int j = 0; j < 16; ++j)
            Bs[bsbase + j * (8 * LDA)] = br[j];
    };

    auto compute = [&](int buf) {
        const float* As = smem + buf * STAGE_F;
        const float* Bs = As + M_BLK * LDA;
        #pragma unroll
        for (int kk = 0; kk < K_C; kk += 4) {
            v2f a[2], b[4];
            #pragma unroll
            for (int mt = 0; mt < 2; ++mt)
                a[mt] = *(const v2f*)&As[afrag + mt * 16 * LDA + kk];
            #pragma unroll
            for (int nt = 0; nt < 4; ++nt)
                b[nt] = *(const v2f*)&Bs[bfrag + nt * 16 * LDA + kk];
            #pragma unroll
            for (int mt = 0; mt < 2; ++mt)
                #pragma unroll
                for (int nt = 0; nt < 4; ++nt)
                    acc[mt][nt] = __builtin_amdgcn_wmma_f32_16x16x4_f32(
                        false, a[mt], false, b[nt],
                        (short)0, acc[mt][nt], false, false);
        }
    };

    // ---- software pipeline over 36 chunks, one barrier per chunk.
    stage_load(0);
    stage_store(0);
    __syncthreads();
    for (int t = 0; t < NCHUNK - 1; ++t) {
        stage_load(t + 1);          // global loads in flight during compute
        compute(t & 1);
        stage_store((t + 1) & 1);   // safe: buffer last read at t-1, fenced below
        __syncthreads();
    }
    compute((NCHUNK - 1) & 1);

    // ---- Epilogue: transpose each wave's 32x64 D tile through LDS so global
    // stores have lanes along M (consecutive q) -> coalesced ~128B runs.
    __syncthreads();
    float* Ds = smem + wid * (32 * LDD);
    #pragma unroll
    for (int mt = 0; mt < 2; ++mt)
        #pragma unroll
        for (int nt = 0; nt < 4; ++nt)
            #pragma unroll
            for (int r = 0; r < 8; ++r)
                Ds[(mt * 16 + r + 8 * hi) * LDD + nt * 16 + ln] = acc[mt][nt][r];
    __syncthreads();

    // lane = local M row (0..31) of this wave's tile
    int m  = m0 + wm + lane;
    int b  = m / PQ;
    int pq = m - b * PQ;
    int p  = pq / W_OUT;
    int q  = pq - p * W_OUT;
    float* orow = out + (((size_t)(b * C_OUT + n0 + wn) * H_OUT + p) * W_OUT + q);
    #pragma unroll
    for (int o = 0; o < 64; ++o)
        orow[(size_t)o * PQ] = Ds[lane * LDD + o];
}

extern "C" void kernel_launch(void* const* d_in, const int* in_sizes, int n_in,
                              void* d_out, int out_size, void* d_ws, size_t ws_size,
                              hipStream_t stream) {
    const float* x = (const float*)d_in[0];   // 32*128*64*64
    const float* w = (const float*)d_in[1];   // 256*128*3*3
    float* out = (float*)d_out;               // 32*256*62*62

    dim3 grid(M_TOTAL / M_BLK, C_OUT / N_BLK);  // 961 x 2
    conv3x3_wmma_f32_kernel<<<grid, 256, 0, stream>>>(x, w, out);
}